// Top2Gate_56453050139304
// MI455X (gfx1250) — compile-verified
//
#include <hip/hip_runtime.h>
#include <hip/hip_bf16.h>
#include <math.h>
#include <stddef.h>

typedef __attribute__((ext_vector_type(2))) float v2f;
typedef __attribute__((ext_vector_type(8))) float v8f;

constexpr int T   = 4096;   // tokens
constexpr int D   = 2048;   // model dim
constexpr int E   = 64;     // experts
constexpr int CAP = 128;    // capacity = 2*ceil(T/E)*1.0

// ---------------------------------------------------------------------------
// Kernel 1: logits[T,E] = x[T,D] @ wg[E,D]^T using V_WMMA_F32_16X16X4_F32.
// One wave per 16-token tile; wave computes 16x64 (4 N-tiles, A reused).
// A 16x4 f32: lane holds M=lane&15, K = k + (lane>>4)*2 + {0,1}  (float2 load)
// B 4x16 f32: lane holds N=lane&15, same K pair                   (float2 load)
// C 16x16 f32: vgpr i, lane L -> C[(L>>4)*8 + i][L&15]
// ---------------------------------------------------------------------------
__global__ __launch_bounds__(32) void k_gemm_logits(const float* __restrict__ x,
                                                    const float* __restrict__ wg,
                                                    float* __restrict__ logits) {
  const int lane = threadIdx.x;            // single wave32
  const int m0   = blockIdx.x * 16;        // token tile base
  const int r    = lane & 15;              // M row (A) / N col (B) for this lane
  const int kg   = (lane >> 4) * 2;        // K sub-offset: 0 or 2

  v8f acc[4] = {};                         // 4 N-tiles of 16

  const float* xrow = x  + (size_t)(m0 + r) * D;
  const float* wr0  = wg + (size_t)(r +  0) * D;
  const float* wr1  = wg + (size_t)(r + 16) * D;
  const float* wr2  = wg + (size_t)(r + 32) * D;
  const float* wr3  = wg + (size_t)(r + 48) * D;

  #pragma unroll 4
  for (int k = 0; k < D; k += 4) {
    v2f a  = *(const v2f*)(xrow + k + kg);
    v2f b0 = *(const v2f*)(wr0  + k + kg);
    v2f b1 = *(const v2f*)(wr1  + k + kg);
    v2f b2 = *(const v2f*)(wr2  + k + kg);
    v2f b3 = *(const v2f*)(wr3  + k + kg);
    acc[0] = __builtin_amdgcn_wmma_f32_16x16x4_f32(false, a, false, b0, (short)0, acc[0], false, false);
    acc[1] = __builtin_amdgcn_wmma_f32_16x16x4_f32(false, a, false, b1, (short)0, acc[1], false, false);
    acc[2] = __builtin_amdgcn_wmma_f32_16x16x4_f32(false, a, false, b2, (short)0, acc[2], false, false);
    acc[3] = __builtin_amdgcn_wmma_f32_16x16x4_f32(false, a, false, b3, (short)0, acc[3], false, false);
  }

  const int mhi = (lane >> 4) * 8;
  #pragma unroll
  for (int j = 0; j < 4; ++j)
    #pragma unroll
    for (int i = 0; i < 8; ++i)
      logits[(size_t)(m0 + mhi + i) * E + j * 16 + r] = acc[j][i];
}

// ---------------------------------------------------------------------------
// Kernel 2: per-token softmax + top1/top2; accumulate me[e] = sum_t gates[t,e]
// ---------------------------------------------------------------------------
__global__ __launch_bounds__(256) void k_gate_top2(const float* __restrict__ logits,
                                                   float* __restrict__ g1,
                                                   float* __restrict__ g2,
                                                   int* __restrict__ idx1,
                                                   int* __restrict__ idx2,
                                                   float* __restrict__ me_sum) {
  __shared__ float sme[E];
  for (int i = threadIdx.x; i < E; i += blockDim.x) sme[i] = 0.0f;
  __syncthreads();

  const int t = blockIdx.x * blockDim.x + threadIdx.x;
  if (t < T) {
    const float* row = logits + (size_t)t * E;
    float m1 = -INFINITY; int i1 = 0;
    #pragma unroll 8
    for (int e = 0; e < E; ++e) {                 // first-index argmax (matches jnp)
      float v = row[e];
      if (v > m1) { m1 = v; i1 = e; }
    }
    float m2 = -INFINITY; int i2 = 0;
    #pragma unroll 8
    for (int e = 0; e < E; ++e) {                 // argmax with top1 masked to -inf
      if (e == i1) continue;
      float v = row[e];
      if (v > m2) { m2 = v; i2 = e; }
    }
    float s = 0.0f;
    #pragma unroll 8
    for (int e = 0; e < E; ++e) s += expf(row[e] - m1);
    const float inv = 1.0f / s;
    g1[t] = inv;                                  // exp(0)*inv
    g2[t] = expf(m2 - m1) * inv;
    idx1[t] = i1;
    idx2[t] = i2;
    #pragma unroll 8
    for (int e = 0; e < E; ++e) atomicAdd(&sme[e], expf(row[e] - m1) * inv);
  }
  __syncthreads();
  for (int i = threadIdx.x; i < E; i += blockDim.x) atomicAdd(&me_sum[i], sme[i]);
}

// ---------------------------------------------------------------------------
// Kernel 3: token-order scan (one lane per expert), capacity positions + l_aux
// loc1[t] = rank of t among tokens with idx1==e ; loc2[t] = rank2 + count1[e]
// ---------------------------------------------------------------------------
__global__ __launch_bounds__(64) void k_scan(const int* __restrict__ idx1,
                                             const int* __restrict__ idx2,
                                             const float* __restrict__ me_sum,
                                             int* __restrict__ loc1,
                                             int* __restrict__ loc2,
                                             float* __restrict__ out /* out[0]=l_aux */) {
  const int e = threadIdx.x;                      // 0..63, two waves
  int cnt1 = 0;
  for (int t = 0; t < T; ++t)
    if (idx1[t] == e) loc1[t] = cnt1++;
  int cnt2 = 0;
  for (int t = 0; t < T; ++t)
    if (idx2[t] == e) loc2[t] = cnt2++ + cnt1;    // offset by total top1 count

  __shared__ float red[E];
  const float me = me_sum[e] * (1.0f / T);
  const float ce = (float)cnt1 * (1.0f / T);
  red[e] = me * ce;
  __syncthreads();
  if (e == 0) {
    float s = 0.0f;
    for (int i = 0; i < E; ++i) s += red[i];
    out[0] = (s / E) * E * E;                     // mean(me*ce)*E*E
  }
}

// ---------------------------------------------------------------------------
// Kernel 4a: zero-fill the 268MB output (b128 stores) — bandwidth floor
// ---------------------------------------------------------------------------
__global__ __launch_bounds__(256) void k_zero(float4* __restrict__ out4, size_t n4,
                                              float* __restrict__ out, size_t n) {
  const size_t i = (size_t)blockIdx.x * blockDim.x + threadIdx.x;
  if (i < n4) {
    float4 z; z.x = 0.f; z.y = 0.f; z.z = 0.f; z.w = 0.f;
    out4[i] = z;
  }
  if (i == 0)
    for (size_t j = n4 * 4; j < n; ++j) out[j] = 0.0f;  // tail (n % 4 elems)
}

// ---------------------------------------------------------------------------
// Kernel 4b: sparse scatter of the 2 nonzero (e,c) slots per token
// ---------------------------------------------------------------------------
__global__ __launch_bounds__(256) void k_scatter(const int* __restrict__ idx1,
                                                 const int* __restrict__ idx2,
                                                 const int* __restrict__ loc1,
                                                 const int* __restrict__ loc2,
                                                 const float* __restrict__ g1,
                                                 const float* __restrict__ g2,
                                                 float* __restrict__ out) {
  const int t = blockIdx.x * blockDim.x + threadIdx.x;
  if (t >= T) return;
  const int  e1 = idx1[t], e2 = idx2[t];
  const int  l1 = loc1[t], l2 = loc2[t];
  const bool keep1 = l1 < CAP, keep2 = l2 < CAP;
  const float G1 = keep1 ? g1[t] : 0.0f;
  const float G2 = keep2 ? g2[t] : 0.0f;
  float denom = G1 + G2;
  denom = fmaxf(denom, 1.1920929e-07f);           // finfo(f32).eps
  const float c1 = G1 / denom;
  const float c2 = G2 / denom;

  float* cw = out + 1;                                   // combine_weights [T,E,CAP]
  float* dm = out + 1 + (size_t)T * E * CAP;             // dispatch_mask  [T,E,CAP]
  if (keep1) {
    const size_t off = ((size_t)t * E + e1) * CAP + l1;
    cw[off] = c1;
    dm[off] = (c1 > 0.0f) ? 1.0f : 0.0f;
  }
  if (keep2) {
    const size_t off = ((size_t)t * E + e2) * CAP + l2;
    cw[off] = c2;
    dm[off] = (c2 > 0.0f) ? 1.0f : 0.0f;
  }
}

__global__ void k_init_me(float* me_sum) {
  if (threadIdx.x < E) me_sum[threadIdx.x] = 0.0f;
}

// ---------------------------------------------------------------------------
extern "C" void kernel_launch(void* const* d_in, const int* in_sizes, int n_in,
                              void* d_out, int out_size, void* d_ws, size_t ws_size,
                              hipStream_t stream) {
  const float* x  = (const float*)d_in[0];   // [T, D]
  const float* wg = (const float*)d_in[1];   // [E, D]
  float* out = (float*)d_out;                // [1 + T*E*CAP + T*E*CAP]

  // workspace layout (~1.15 MB)
  char* ws = (char*)d_ws;
  float* logits = (float*)ws;                 ws += (size_t)T * E * sizeof(float);
  float* g1     = (float*)ws;                 ws += (size_t)T * sizeof(float);
  float* g2     = (float*)ws;                 ws += (size_t)T * sizeof(float);
  int*   idx1   = (int*)ws;                   ws += (size_t)T * sizeof(int);
  int*   idx2   = (int*)ws;                   ws += (size_t)T * sizeof(int);
  int*   loc1   = (int*)ws;                   ws += (size_t)T * sizeof(int);
  int*   loc2   = (int*)ws;                   ws += (size_t)T * sizeof(int);
  float* me_sum = (float*)ws;                 ws += (size_t)E * sizeof(float);

  const size_t n  = (size_t)out_size;
  const size_t n4 = n >> 2;

  // Zero the (poisoned) output: dominant cost, pure store bandwidth.
  k_zero<<<dim3((unsigned)((n4 + 255) / 256)), dim3(256), 0, stream>>>(
      (float4*)out, n4, out, n);
  k_init_me<<<dim3(1), dim3(64), 0, stream>>>(me_sum);

  // Logits via f32 WMMA: one wave per 16-token tile.
  k_gemm_logits<<<dim3(T / 16), dim3(32), 0, stream>>>(x, wg, logits);

  // Softmax + top-2 per token.
  k_gate_top2<<<dim3(T / 256), dim3(256), 0, stream>>>(logits, g1, g2, idx1, idx2, me_sum);

  // Serial token-order scan (expert-per-lane) + l_aux.
  k_scan<<<dim3(1), dim3(64), 0, stream>>>(idx1, idx2, me_sum, loc1, loc2, out);

  // Sparse scatter of combine_weights / dispatch_mask.
  k_scatter<<<dim3(T / 256), dim3(256), 0, stream>>>(idx1, idx2, loc1, loc2, g1, g2, out);
}